// GCN_57775900066214
// MI455X (gfx1250) — compile-verified
//
#include <hip/hip_runtime.h>
#include <stdint.h>

typedef __attribute__((ext_vector_type(2))) float v2f;
typedef __attribute__((ext_vector_type(8))) float v8f;

#define D 64

// Native f32 global atomic add (no-return form -> STOREcnt), device scope so
// cross-workgroup accumulation is coherent in L2.
__device__ __forceinline__ void atomic_add_f32(float* p, float v) {
    asm volatile("global_atomic_add_f32 %0, %1, off scope:SCOPE_DEV"
                 :
                 : "v"((uint64_t)(uintptr_t)p), "v"(v)
                 : "memory");
}

// out = 0, deg = 2 (each self-loop edge (k,k) bumps deg[k] twice in the reference)
__global__ void gcn_init(float* __restrict__ out, float* __restrict__ deg, int n_nodes) {
    int idx = blockIdx.x * blockDim.x + threadIdx.x;
    int total = n_nodes * D;
    if (idx < total) out[idx] = 0.0f;
    if (idx < n_nodes) deg[idx] = 2.0f;
}

__global__ void gcn_degree(const int* __restrict__ src, const int* __restrict__ dst,
                           float* __restrict__ deg, int n_edges) {
    int e = blockIdx.x * blockDim.x + threadIdx.x;
    if (e >= n_edges) return;
    atomic_add_f32(&deg[src[e]], 1.0f);
    atomic_add_f32(&deg[dst[e]], 1.0f);
}

// H = X @ W^T + b using V_WMMA_F32_16X16X4_F32 (one 16x16 tile per wave).
__global__ __launch_bounds__(256) void gcn_gemm(const float* __restrict__ X,
                                                const float* __restrict__ W,
                                                const float* __restrict__ bias,
                                                float* __restrict__ H, int n_nodes) {
    const int lane = threadIdx.x & 31;
    const int wid  = threadIdx.x >> 5;   // 0..7
    const int row_tile = wid >> 2;       // 0..1
    const int col_tile = wid & 3;        // 0..3
    const int row_base = blockIdx.x * 32 + row_tile * 16;
    const int n0 = col_tile * 16;
    if (row_base >= n_nodes) return;

    const int l15   = lane & 15;
    const int khalf = (lane >> 4) * 2;   // lanes 0-15 -> K={0,1}; lanes 16-31 -> K={2,3}

    int arow = row_base + l15;
    if (arow >= n_nodes) arow = n_nodes - 1;            // clamp (stores are guarded)
    const float* aptr = X + (size_t)arow * D + khalf;   // A[m][k] = X[row][k]
    const float* bptr = W + (size_t)(n0 + l15) * D + khalf; // B[k][n] = W^T[k][n] = W[n][k]

    v8f c = {0.f, 0.f, 0.f, 0.f, 0.f, 0.f, 0.f, 0.f};
#pragma unroll
    for (int k0 = 0; k0 < D; k0 += 4) {
        v2f a, b;
        a.x = aptr[k0];
        a.y = aptr[k0 + 1];
        b.x = bptr[k0];
        b.y = bptr[k0 + 1];
        // 8 args: (neg_a, A, neg_b, B, c_mod, C, reuse_a, reuse_b)
        c = __builtin_amdgcn_wmma_f32_16x16x4_f32(false, a, false, b, (short)0, c,
                                                  false, false);
    }

    const float bv = bias[n0 + l15];       // bias depends only on the column
    const int roff = (lane >> 4) * 8;      // lanes 16-31 hold rows M=8..15
#pragma unroll
    for (int v = 0; v < 8; ++v) {
        int row = row_base + v + roff;
        if (row < n_nodes) H[(size_t)row * D + n0 + l15] = c[v] + bv;
    }
}

// One wave per edge: out[src] += norm * H[dst]; 2 features per lane.
// H + out + deg (~52 MB) are L2-resident (192 MB), so this runs at L2 speed.
__global__ __launch_bounds__(256) void gcn_scatter(const int* __restrict__ src,
                                                   const int* __restrict__ dst,
                                                   const float* __restrict__ H,
                                                   const float* __restrict__ deg,
                                                   float* __restrict__ out, int n_edges) {
    int e = (int)((blockIdx.x * blockDim.x + threadIdx.x) >> 5);
    if (e >= n_edges) return;
    const int lane = threadIdx.x & 31;
    const int s = src[e];
    const int d = dst[e];
    const float norm = rsqrtf(deg[s] * deg[d]);
    const float* hrow = H + (size_t)d * D;
    float* orow = out + (size_t)s * D;
    atomic_add_f32(&orow[lane],      hrow[lane]      * norm);
    atomic_add_f32(&orow[lane + 32], hrow[lane + 32] * norm);
}

// Add self-loop term (norm = 1/deg) and apply ReLU.
__global__ void gcn_finish(float* __restrict__ out, const float* __restrict__ H,
                           const float* __restrict__ deg, int n_nodes) {
    int idx = blockIdx.x * blockDim.x + threadIdx.x;
    if (idx >= n_nodes * D) return;
    float inv = 1.0f / deg[idx >> 6];
    float v = out[idx] + H[idx] * inv;
    out[idx] = v > 0.0f ? v : 0.0f;
}

extern "C" void kernel_launch(void* const* d_in, const int* in_sizes, int n_in,
                              void* d_out, int out_size, void* d_ws, size_t ws_size,
                              hipStream_t stream) {
    const float* X    = (const float*)d_in[0];  // [N, 64]
    const int*   edg  = (const int*)d_in[1];    // [2, E] row-major
    const float* W    = (const float*)d_in[2];  // [64, 64]
    const float* bias = (const float*)d_in[3];  // [64]
    float* out = (float*)d_out;                 // [N, 64]

    const int n_nodes = in_sizes[0] / D;
    const int n_edges = in_sizes[1] / 2;
    const int* src = edg;
    const int* dst = edg + n_edges;

    // Workspace layout: deg [N] f32 | H [N*64] f32
    float* deg = (float*)d_ws;
    size_t deg_bytes = (((size_t)n_nodes * sizeof(float)) + 255) & ~(size_t)255;
    float* H = (float*)((char*)d_ws + deg_bytes);

    const int total = n_nodes * D;
    gcn_init   <<<(total + 255) / 256, 256, 0, stream>>>(out, deg, n_nodes);
    gcn_degree <<<(n_edges + 255) / 256, 256, 0, stream>>>(src, dst, deg, n_edges);
    gcn_gemm   <<<(n_nodes + 31) / 32, 256, 0, stream>>>(X, W, bias, H, n_nodes);
    gcn_scatter<<<(n_edges + 7) / 8, 256, 0, stream>>>(src, dst, H, deg, out, n_edges);
    gcn_finish <<<(total + 255) / 256, 256, 0, stream>>>(out, H, deg, n_nodes);
}